// GraphAttentionLayer_22170621182160
// MI455X (gfx1250) — compile-verified
//
#include <hip/hip_runtime.h>
#include <math.h>

typedef __attribute__((ext_vector_type(2))) float v2f;
typedef __attribute__((ext_vector_type(8))) float v8f;

#define BS    32
#define NN    1024   // N1 == N2
#define FF    1024   // F_IN == F_OUT
#define ALPHA 0.2f

#define BM 128
#define BN 128
#define BK 32
#define LDX 129      // k-major transposed tile: stride ≡ 1 mod 64 -> conflict-free scatter fill
#define LDW 132      // row tile: 16B-aligned rows for ds_store_b128
#define THREADS 256  // 8 waves (wave32)

// ---------------------------------------------------------------------------
__global__ void zero_f32(float* __restrict__ p, int n) {
    int i = blockIdx.x * blockDim.x + threadIdx.x;
    if (i < n) p[i] = 0.0f;
}

// ---------------------------------------------------------------------------
// Stage 1: Y[b] = X[b] @ W  (fp32 WMMA 16x16x4, reg-double-buffered LDS tiles),
// fused epilogue: s[b,m] += Y[b,m,:]·avec  (atomic, s pre-zeroed)
// grid = (FF/BN, NN/BM, BS), block = 256
// ---------------------------------------------------------------------------
__global__ __launch_bounds__(THREADS)
void gemm_xw_s(const float* __restrict__ X,     // [BS, NN, FF]
               const float* __restrict__ W,     // [FF, FF]
               const float* __restrict__ avec,  // [FF]
               float* __restrict__ Y,           // [BS, NN, FF]
               float* __restrict__ s)           // [BS, NN]
{
    __shared__ float Xs[BK][LDX];   // k-major (transposed) X tile
    __shared__ float Wt[BK][LDW];   // k-major W tile (row-contiguous in n)

    const int b    = blockIdx.z;
    const int row0 = blockIdx.y * BM;
    const int col0 = blockIdx.x * BN;

    const int tid   = threadIdx.x;
    const int wave  = tid >> 5;
    const int lane  = tid & 31;
    const int lhalf = lane >> 4;
    const int l16   = lane & 15;
    const int wm    = wave & 3;      // 4 waves along M (32 rows each)
    const int wn    = wave >> 2;     // 2 waves along N (64 cols each)

    const float* Xb = X + (size_t)b * NN * FF;

    // fill mappings (float4 granularity)
    const int xq = tid & 7,  xm = tid >> 3;   // X: 8 float4 per row,  rows step 32/iter
    const int wq = tid & 31, wk = tid >> 5;   // W: 32 float4 per row, k rows step 8/iter

    v8f acc[2][4];
#pragma unroll
    for (int tm = 0; tm < 2; ++tm)
#pragma unroll
        for (int tn = 0; tn < 4; ++tn) acc[tm][tn] = (v8f)0.0f;

    float4 xr[4], wr[4];
    // prologue: stage tile k0 = 0 into registers
#pragma unroll
    for (int i = 0; i < 4; ++i) {
        xr[i] = *(const float4*)&Xb[(size_t)(row0 + xm + i * 32) * FF + (xq * 4)];
        wr[i] = *(const float4*)&W[(size_t)(wk + i * 8) * FF + (col0 + wq * 4)];
    }

    for (int k0 = 0; k0 < FF; k0 += BK) {
        __syncthreads();   // previous compute done reading LDS
        // regs -> LDS
#pragma unroll
        for (int i = 0; i < 4; ++i) {
            Xs[xq * 4 + 0][xm + i * 32] = xr[i].x;
            Xs[xq * 4 + 1][xm + i * 32] = xr[i].y;
            Xs[xq * 4 + 2][xm + i * 32] = xr[i].z;
            Xs[xq * 4 + 3][xm + i * 32] = xr[i].w;
            *(float4*)&Wt[wk + i * 8][wq * 4] = wr[i];
        }
        __syncthreads();
        // stage NEXT tile into registers: latency overlapped with WMMA loop below
        if (k0 + BK < FF) {
            int kn = k0 + BK;
#pragma unroll
            for (int i = 0; i < 4; ++i) {
                xr[i] = *(const float4*)&Xb[(size_t)(row0 + xm + i * 32) * FF + (kn + xq * 4)];
                wr[i] = *(const float4*)&W[(size_t)(kn + wk + i * 8) * FF + (col0 + wq * 4)];
            }
            if (kn + BK < FF)   // warm L2/L0 for the tile after next (global_prefetch_b8)
                __builtin_prefetch(&Xb[(size_t)(row0 + xm) * FF + (kn + BK)], 0, 1);
        }

#pragma unroll
        for (int kk = 0; kk < BK; kk += 4) {
            v2f afrag[2], bfrag[4];
#pragma unroll
            for (int tm = 0; tm < 2; ++tm) {
                int m = wm * 32 + tm * 16 + l16;
                afrag[tm].x = Xs[kk + lhalf * 2 + 0][m];
                afrag[tm].y = Xs[kk + lhalf * 2 + 1][m];
            }
#pragma unroll
            for (int tn = 0; tn < 4; ++tn) {
                int n = wn * 64 + tn * 16 + l16;
                bfrag[tn].x = Wt[kk + lhalf * 2 + 0][n];
                bfrag[tn].y = Wt[kk + lhalf * 2 + 1][n];
            }
#pragma unroll
            for (int tm = 0; tm < 2; ++tm)
#pragma unroll
                for (int tn = 0; tn < 4; ++tn)
                    acc[tm][tn] = __builtin_amdgcn_wmma_f32_16x16x4_f32(
                        false, afrag[tm], false, bfrag[tn],
                        (short)0, acc[tm][tn], false, false);
        }
    }

    // ---- epilogue: store Y tile, fold tile·avec into s via shfl + atomics ----
    float* Yb = Y + (size_t)b * NN * FF;
    float a_l[4];
#pragma unroll
    for (int tn = 0; tn < 4; ++tn) a_l[tn] = avec[col0 + wn * 64 + tn * 16 + l16];

#pragma unroll
    for (int tm = 0; tm < 2; ++tm) {
#pragma unroll
        for (int r = 0; r < 8; ++r) {
            int m = row0 + wm * 32 + tm * 16 + lhalf * 8 + r;
            float ssum = 0.0f;
#pragma unroll
            for (int tn = 0; tn < 4; ++tn) {
                float c = acc[tm][tn][r];
                Yb[(size_t)m * FF + (col0 + wn * 64 + tn * 16 + l16)] = c;
                ssum += c * a_l[tn];
            }
#pragma unroll
            for (int off = 1; off < 16; off <<= 1)
                ssum += __shfl_xor(ssum, off, 32);
            if (l16 == 0) atomicAdd(&s[(size_t)b * NN + m], ssum);
        }
    }
}

// ---------------------------------------------------------------------------
// Stage 2: per (b,m): softmax over n of leaky(s1[n]+s2[m]); write attT[b,m,n].
// One wave per column; float4 sweeps over s1.
// ---------------------------------------------------------------------------
__global__ __launch_bounds__(256)
void softmax_attT(const float* __restrict__ s1,
                  const float* __restrict__ s2,
                  float* __restrict__ attT)     // [BS, NN(m), NN(n)]
{
    int gwave = (blockIdx.x * blockDim.x + threadIdx.x) >> 5;
    int lane  = threadIdx.x & 31;
    int b = gwave >> 10;
    int m = gwave & (NN - 1);
    if (b >= BS) return;

    const float   s2m  = s2[(size_t)b * NN + m];
    const float4* s1b4 = (const float4*)(s1 + (size_t)b * NN);

    float mx = -3.402823466e+38f;
#pragma unroll
    for (int it = 0; it < NN / 128; ++it) {
        float4 v = s1b4[it * 32 + lane];
        float e0 = v.x + s2m; e0 = e0 > 0.0f ? e0 : ALPHA * e0;
        float e1 = v.y + s2m; e1 = e1 > 0.0f ? e1 : ALPHA * e1;
        float e2 = v.z + s2m; e2 = e2 > 0.0f ? e2 : ALPHA * e2;
        float e3 = v.w + s2m; e3 = e3 > 0.0f ? e3 : ALPHA * e3;
        mx = fmaxf(mx, fmaxf(fmaxf(e0, e1), fmaxf(e2, e3)));
    }
#pragma unroll
    for (int off = 1; off < 32; off <<= 1) mx = fmaxf(mx, __shfl_xor(mx, off, 32));

    float sum = 0.0f;
#pragma unroll
    for (int it = 0; it < NN / 128; ++it) {
        float4 v = s1b4[it * 32 + lane];
        float e0 = v.x + s2m; e0 = e0 > 0.0f ? e0 : ALPHA * e0;
        float e1 = v.y + s2m; e1 = e1 > 0.0f ? e1 : ALPHA * e1;
        float e2 = v.z + s2m; e2 = e2 > 0.0f ? e2 : ALPHA * e2;
        float e3 = v.w + s2m; e3 = e3 > 0.0f ? e3 : ALPHA * e3;
        sum += __expf(e0 - mx) + __expf(e1 - mx) + __expf(e2 - mx) + __expf(e3 - mx);
    }
#pragma unroll
    for (int off = 1; off < 32; off <<= 1) sum += __shfl_xor(sum, off, 32);
    float inv = 1.0f / sum;

    float4* A = (float4*)(attT + ((size_t)b * NN + m) * NN);
#pragma unroll
    for (int it = 0; it < NN / 128; ++it) {
        float4 v = s1b4[it * 32 + lane];
        float e0 = v.x + s2m; e0 = e0 > 0.0f ? e0 : ALPHA * e0;
        float e1 = v.y + s2m; e1 = e1 > 0.0f ? e1 : ALPHA * e1;
        float e2 = v.z + s2m; e2 = e2 > 0.0f ? e2 : ALPHA * e2;
        float e3 = v.w + s2m; e3 = e3 > 0.0f ? e3 : ALPHA * e3;
        float4 o;
        o.x = __expf(e0 - mx) * inv;
        o.y = __expf(e1 - mx) * inv;
        o.z = __expf(e2 - mx) * inv;
        o.w = __expf(e3 - mx) * inv;
        A[it * 32 + lane] = o;
    }
}

// ---------------------------------------------------------------------------
// Stage 3: out[b] = elu( attT[b] @ Ws[b] + We[b] )  (same WMMA pipeline)
// ---------------------------------------------------------------------------
__global__ __launch_bounds__(THREADS)
void gemm_att(const float* __restrict__ AT,   // [BS, NN(m), NN(n)]
              const float* __restrict__ Ws,   // [BS, NN(n), FF(o)]
              const float* __restrict__ We,   // [BS, NN(m), FF(o)]
              float* __restrict__ out)        // [BS, NN(m), FF(o)]
{
    __shared__ float As[BK][LDX];
    __shared__ float Bt[BK][LDW];

    const int b    = blockIdx.z;
    const int row0 = blockIdx.y * BM;   // m
    const int col0 = blockIdx.x * BN;   // o

    const int tid   = threadIdx.x;
    const int wave  = tid >> 5;
    const int lane  = tid & 31;
    const int lhalf = lane >> 4;
    const int l16   = lane & 15;
    const int wm    = wave & 3;
    const int wn    = wave >> 2;

    const float* Ab = AT + (size_t)b * NN * NN;
    const float* Bb = Ws + (size_t)b * NN * FF;

    const int xq = tid & 7,  xm = tid >> 3;
    const int wq = tid & 31, wk = tid >> 5;

    v8f acc[2][4];
#pragma unroll
    for (int tm = 0; tm < 2; ++tm)
#pragma unroll
        for (int tn = 0; tn < 4; ++tn) acc[tm][tn] = (v8f)0.0f;

    float4 xr[4], wr[4];
#pragma unroll
    for (int i = 0; i < 4; ++i) {
        xr[i] = *(const float4*)&Ab[(size_t)(row0 + xm + i * 32) * NN + (xq * 4)];
        wr[i] = *(const float4*)&Bb[(size_t)(wk + i * 8) * FF + (col0 + wq * 4)];
    }

    for (int k0 = 0; k0 < NN; k0 += BK) {
        __syncthreads();
#pragma unroll
        for (int i = 0; i < 4; ++i) {
            As[xq * 4 + 0][xm + i * 32] = xr[i].x;
            As[xq * 4 + 1][xm + i * 32] = xr[i].y;
            As[xq * 4 + 2][xm + i * 32] = xr[i].z;
            As[xq * 4 + 3][xm + i * 32] = xr[i].w;
            *(float4*)&Bt[wk + i * 8][wq * 4] = wr[i];
        }
        __syncthreads();
        if (k0 + BK < NN) {
            int kn = k0 + BK;
#pragma unroll
            for (int i = 0; i < 4; ++i) {
                xr[i] = *(const float4*)&Ab[(size_t)(row0 + xm + i * 32) * NN + (kn + xq * 4)];
                wr[i] = *(const float4*)&Bb[(size_t)(kn + wk + i * 8) * FF + (col0 + wq * 4)];
            }
            if (kn + BK < NN)
                __builtin_prefetch(&Ab[(size_t)(row0 + xm) * NN + (kn + BK)], 0, 1);
        }

#pragma unroll
        for (int kk = 0; kk < BK; kk += 4) {
            v2f afrag[2], bfrag[4];
#pragma unroll
            for (int tm = 0; tm < 2; ++tm) {
                int m = wm * 32 + tm * 16 + l16;
                afrag[tm].x = As[kk + lhalf * 2 + 0][m];
                afrag[tm].y = As[kk + lhalf * 2 + 1][m];
            }
#pragma unroll
            for (int tn = 0; tn < 4; ++tn) {
                int n = wn * 64 + tn * 16 + l16;
                bfrag[tn].x = Bt[kk + lhalf * 2 + 0][n];
                bfrag[tn].y = Bt[kk + lhalf * 2 + 1][n];
            }
#pragma unroll
            for (int tm = 0; tm < 2; ++tm)
#pragma unroll
                for (int tn = 0; tn < 4; ++tn)
                    acc[tm][tn] = __builtin_amdgcn_wmma_f32_16x16x4_f32(
                        false, afrag[tm], false, bfrag[tn],
                        (short)0, acc[tm][tn], false, false);
        }
    }

    // ---- epilogue: h = acc + We ; out = h>0 ? h : expm1(h)  (streaming NT store)
    const float* Web = We + (size_t)b * NN * FF;
    float* Ob = out + (size_t)b * NN * FF;
#pragma unroll
    for (int tm = 0; tm < 2; ++tm) {
#pragma unroll
        for (int r = 0; r < 8; ++r) {
            int m = row0 + wm * 32 + tm * 16 + lhalf * 8 + r;
#pragma unroll
            for (int tn = 0; tn < 4; ++tn) {
                int o = col0 + wn * 64 + tn * 16 + l16;
                float h = acc[tm][tn][r] + Web[(size_t)m * FF + o];
                h = h > 0.0f ? h : expm1f(h);
                __builtin_nontemporal_store(h, &Ob[(size_t)m * FF + o]);
            }
        }
    }
}

// ---------------------------------------------------------------------------
extern "C" void kernel_launch(void* const* d_in, const int* in_sizes, int n_in,
                              void* d_out, int out_size, void* d_ws, size_t ws_size,
                              hipStream_t stream) {
    const float* X   = (const float*)d_in[0];  // start_feature [BS,NN,FF]
    const float* E   = (const float*)d_in[1];  // end_feature   [BS,NN,FF]
    const float* Wsp = (const float*)d_in[2];  // Ws_param [FF,FF]
    const float* Wep = (const float*)d_in[3];  // We_param [FF,FF]
    const float* a   = (const float*)d_in[4];  // a_param [2*FF]
    float* out = (float*)d_out;

    float* ws   = (float*)d_ws;
    size_t big  = (size_t)BS * NN * FF;
    float* Wsb  = ws;                          // Ws       (re-read in stage 3 -> temporal)
    float* Web  = Wsb + big;                   // We
    float* attT = Web + big;                   // attention, transposed [b,m,n]
    float* s1   = attT + big;
    float* s2   = s1 + (size_t)BS * NN;

    zero_f32<<<(2 * BS * NN + 255) / 256, 256, 0, stream>>>(s1, 2 * BS * NN);

    dim3 grid(FF / BN, NN / BM, BS);           // (8, 8, 32)
    gemm_xw_s<<<grid, THREADS, 0, stream>>>(X, Wsp, a,      Wsb, s1);
    gemm_xw_s<<<grid, THREADS, 0, stream>>>(E, Wep, a + FF, Web, s2);

    softmax_attT<<<(BS * NN) / 8, 256, 0, stream>>>(s1, s2, attT);

    gemm_att<<<grid, THREADS, 0, stream>>>(attT, Wsb, Web, out);
}